// SSIM3D_33792802685702
// MI455X (gfx1250) — compile-verified
//
#include <hip/hip_runtime.h>

typedef __attribute__((ext_vector_type(2))) float v2f;
typedef __attribute__((ext_vector_type(8))) float v8f;

// ---------------- problem constants ----------------
constexpr int Din = 128, Hin = 192, Win = 192, NB = 4;
constexpr int KW  = 11;                 // gaussian taps
constexpr int OD  = Din - (KW - 1);     // 118
constexpr int OH  = Hin - (KW - 1);     // 182
constexpr int OW  = Win - (KW - 1);     // 182
constexpr int TZ  = (OD + 15) / 16;     // 8
constexpr int TY  = (OH + 15) / 16;     // 12
constexpr int TX  = (OW + 15) / 16;     // 12
constexpr int NPART = TX * TY * TZ * NB; // 4608 partial sums

// ---------------- LDS layout (floats) ----------------
constexpr int IN_ELEMS = 26 * 26 * 26;  // 17576 halo tile
constexpr int IN_PAD   = 17920;         // rows 676..687 + K-overrun pad, zeroed
constexpr int ROWG     = 43;            // ceil(676/16) row groups for W-pass
constexpr int TMPW_OFF = 2 * IN_PAD;                 // 35840
constexpr int TMPW_SZ  = ROWG * 16 * 16;             // 11008 (688 rows x 16)
constexpr int TMPH_OFF = TMPW_OFF + TMPW_SZ;         // 46848
constexpr int TMPH_SZ  = 28 * 16 * 16;               // 7168 (2 zero pad K-rows)
constexpr int FIN_OFF  = TMPH_OFF + TMPH_SZ;         // 54016
constexpr int SMEM_FLOATS = FIN_OFF + 5 * 4096;      // 74496 -> 297984 B

__device__ __forceinline__ v8f wmma_f32(v2f a, v2f b, v8f c) {
  return __builtin_amdgcn_wmma_f32_16x16x4_f32(false, a, false, b,
                                               (short)0, c, false, false);
}

// Per-quantity separable conv pipeline, fully compile-time specialized.
// Q: 0:i1  1:i2  2:i1*i1  3:i2*i2  4:i1*i2
template <int Q>
__device__ __forceinline__ void conv_q(const float* __restrict__ lds1,
                                       const float* __restrict__ lds2,
                                       float* __restrict__ tmpW,
                                       float* __restrict__ tmpH,
                                       float* __restrict__ qbuf,
                                       const v2f* aC, int wv, int lh, int nL) {
  // ---- W-pass: rows = (d,h) pairs (676, padded to 688), conv along x ----
  for (int g = wv; g < ROWG; g += 8) {
    int r0 = g * 16;
    const float* p1 = lds1 + (r0 + nL) * 26;
    const float* p2 = lds2 + (r0 + nL) * 26;
    v8f acc = {};
#pragma unroll
    for (int c = 0; c < 7; ++c) {
      int kk = 4 * c + 2 * lh;       // unguarded: k=26,27 hit zeroed pad, A=0
      v2f b;
      if (Q == 0) {
        b.x = p1[kk]; b.y = p1[kk + 1];
      } else if (Q == 1) {
        b.x = p2[kk]; b.y = p2[kk + 1];
      } else if (Q == 2) {
        float a0 = p1[kk], a1 = p1[kk + 1];
        b.x = a0 * a0; b.y = a1 * a1;
      } else if (Q == 3) {
        float a0 = p2[kk], a1 = p2[kk + 1];
        b.x = a0 * a0; b.y = a1 * a1;
      } else {
        b.x = p1[kk] * p2[kk]; b.y = p1[kk + 1] * p2[kk + 1];
      }
      acc = wmma_f32(aC[c], b, acc);
    }
    float* dst = tmpW + (r0 + nL) * 16;
#pragma unroll
    for (int i = 0; i < 8; ++i) dst[i + 8 * lh] = acc[i];
  }
  __syncthreads();

  // ---- H-pass: per d-plane (26), conv along h ----
  // B[k][n] = tmpW[(d*26+k)*16+n]; k=26,27 read rows 676/677 (exact zeros)
  for (int d = wv; d < 26; d += 8) {
    const float* bp = tmpW + d * 26 * 16 + nL;
    v8f acc = {};
#pragma unroll
    for (int c = 0; c < 7; ++c) {
      int kk = 4 * c + 2 * lh;
      v2f b; b.x = bp[kk * 16]; b.y = bp[(kk + 1) * 16];
      acc = wmma_f32(aC[c], b, acc);
    }
    float* dst = tmpH + d * 256 + nL;
#pragma unroll
    for (int i = 0; i < 8; ++i) dst[(i + 8 * lh) * 16] = acc[i];
  }
  __syncthreads();

  // ---- D-pass: per y (16), conv along d ----
  // B[k][n] = tmpH[k*256 + y*16 + n]; k=26,27 hit zeroed tmpH pad rows
  for (int y = wv; y < 16; y += 8) {
    const float* bp = tmpH + y * 16 + nL;
    v8f acc = {};
#pragma unroll
    for (int c = 0; c < 7; ++c) {
      int kk = 4 * c + 2 * lh;
      v2f b; b.x = bp[kk * 256]; b.y = bp[(kk + 1) * 256];
      acc = wmma_f32(aC[c], b, acc);
    }
    float* dst = qbuf + y * 16 + nL;
#pragma unroll
    for (int i = 0; i < 8; ++i) dst[(i + 8 * lh) * 256] = acc[i];
  }
  __syncthreads();
}

__global__ void __launch_bounds__(256, 1)
ssim3d_tile(const float* __restrict__ img1, const float* __restrict__ img2,
            float* __restrict__ part) {
  extern __shared__ float smem[];
  __shared__ float wlds[16];
  __shared__ float red[256];

  float* lds1 = smem;
  float* lds2 = smem + IN_PAD;
  float* tmpW = smem + TMPW_OFF;   // [688 rows][16 x_out]
  float* tmpH = smem + TMPH_OFF;   // [28 d][16 y][16 x] (last 2 d = zero pad)
  float* fin  = smem + FIN_OFF;    // 5 x [16 z][16 y][16 x]

  const int tid  = threadIdx.x;
  const int wv   = tid >> 5;       // wave id (0..7), wave32
  const int lane = tid & 31;
  const int lh   = lane >> 4;      // lane half (K-pair select)
  const int nL   = lane & 15;      // column n (== M index for A lanes)

  const int b   = blockIdx.z >> 3;        // batch
  const int tz  = blockIdx.z & 7;         // z tile
  const int zt0 = tz * 16, yt0 = blockIdx.y * 16, xt0 = blockIdx.x * 16;

  // ---- normalized gaussian taps into LDS (wlds[11..15] = 0) ----
  if (tid < 16) {
    float sum = 0.f, g = 0.f;
#pragma unroll
    for (int i = 0; i < 11; ++i) {
      float t = (float)(i - 5);
      float e = __expf(-(t * t) / (2.f * 1.5f * 1.5f));
      sum += e;
      if (i == tid) g = e;
    }
    wlds[tid] = (tid < 11) ? g / (sum + 1e-12f) : 0.f;
  }

  // ---- load 26^3 halo tiles of both volumes (clamped at borders) ----
  const float* i1b = img1 + (size_t)b * Din * Hin * Win;
  const float* i2b = img2 + (size_t)b * Din * Hin * Win;
  for (int li = tid; li < IN_ELEMS; li += 256) {
    int dx = li % 26; int t = li / 26; int dy = t % 26; int dz = t / 26;
    int gz = min(zt0 + dz, Din - 1);
    int gy = min(yt0 + dy, Hin - 1);
    int gx = min(xt0 + dx, Win - 1);
    size_t off = ((size_t)gz * Hin + gy) * Win + gx;
    lds1[li] = i1b[off];
    lds2[li] = i2b[off];
  }
  for (int li = IN_ELEMS + tid; li < IN_PAD; li += 256) {
    lds1[li] = 0.f; lds2[li] = 0.f;
  }
  // zero the two tmpH K-overrun pad rows (read by unguarded D-pass loads)
  for (int li = tid; li < 2 * 256; li += 256) tmpH[26 * 256 + li] = 0.f;
  __syncthreads();

  // ---- per-lane banded-A chunks (A[m][k] = w[k-m], k in [m, m+10]) ----
  // 16x16x4 f32 A layout: lanes 0-15 hold K = 4c+{0,1}, lanes 16-31 K = 4c+{2,3}
  v2f aC[7];
#pragma unroll
  for (int c = 0; c < 7; ++c) {
    int kk = 4 * c + 2 * lh;
    int i0 = kk - nL, i1 = kk + 1 - nL;
    int i0c = min(max(i0, 0), 15), i1c = min(max(i1, 0), 15);
    float a0 = (i0 >= 0 && i0 < 11) ? wlds[i0c] : 0.f;
    float a1 = (i1 >= 0 && i1 < 11) ? wlds[i1c] : 0.f;
    aC[c].x = a0; aC[c].y = a1;
  }

  // ---- 5 quantities through separable 3-pass conv, all via f32 WMMA ----
  conv_q<0>(lds1, lds2, tmpW, tmpH, fin + 0 * 4096, aC, wv, lh, nL);
  conv_q<1>(lds1, lds2, tmpW, tmpH, fin + 1 * 4096, aC, wv, lh, nL);
  conv_q<2>(lds1, lds2, tmpW, tmpH, fin + 2 * 4096, aC, wv, lh, nL);
  conv_q<3>(lds1, lds2, tmpW, tmpH, fin + 3 * 4096, aC, wv, lh, nL);
  conv_q<4>(lds1, lds2, tmpW, tmpH, fin + 4 * 4096, aC, wv, lh, nL);

  // ---- SSIM map + in-block reduction ----
  const float C1 = 1e-4f, C2 = 9e-4f;
  const float* f0 = fin;              // mu1
  const float* f1 = fin + 4096;       // mu2
  const float* f2 = fin + 2 * 4096;   // E[x1^2]
  const float* f3 = fin + 3 * 4096;   // E[x2^2]
  const float* f4 = fin + 4 * 4096;   // E[x1*x2]
  float lsum = 0.f;
  for (int e = tid; e < 4096; e += 256) {
    int zz = e >> 8, yy = (e >> 4) & 15, xx = e & 15;
    if ((zt0 + zz < OD) && (yt0 + yy < OH) && (xt0 + xx < OW)) {
      float mu1 = f0[e], mu2 = f1[e];
      float m11 = mu1 * mu1, m22 = mu2 * mu2, m12 = mu1 * mu2;
      float s11 = f2[e] - m11, s22 = f3[e] - m22, s12 = f4[e] - m12;
      float v1 = 2.f * s12 + C2, v2 = s11 + s22 + C2;
      float num = (2.f * m12 + C1) * v1;
      float den = (m11 + m22 + C1) * v2;
      lsum += num / den;
    }
  }
  red[tid] = lsum;
  __syncthreads();
  for (int s = 128; s > 0; s >>= 1) {
    if (tid < s) red[tid] += red[tid + s];
    __syncthreads();
  }
  if (tid == 0) {
    int wgId = (blockIdx.z * gridDim.y + blockIdx.y) * gridDim.x + blockIdx.x;
    part[wgId] = red[0];
  }
}

// deterministic final reduction (fixed accumulation order, no atomics)
__global__ void ssim3d_reduce(const float* __restrict__ part,
                              float* __restrict__ out) {
  __shared__ float red[256];
  int tid = threadIdx.x;
  float a = 0.f;
  for (int i = tid; i < NPART; i += 256) a += part[i];
  red[tid] = a;
  __syncthreads();
  for (int s = 128; s > 0; s >>= 1) {
    if (tid < s) red[tid] += red[tid + s];
    __syncthreads();
  }
  if (tid == 0)
    out[0] = red[0] / (float)((long long)NB * OD * OH * OW);
}

extern "C" void kernel_launch(void* const* d_in, const int* in_sizes, int n_in,
                              void* d_out, int out_size, void* d_ws,
                              size_t ws_size, hipStream_t stream) {
  (void)in_sizes; (void)n_in; (void)out_size; (void)ws_size;
  const float* img1 = (const float*)d_in[0];
  const float* img2 = (const float*)d_in[1];
  float* part = (float*)d_ws;            // NPART floats of scratch
  float* out  = (float*)d_out;

  size_t shmem = (size_t)SMEM_FLOATS * sizeof(float);  // ~298 KB (<= 320 KB WGP LDS)
  hipFuncSetAttribute(reinterpret_cast<const void*>(ssim3d_tile),
                      hipFuncAttributeMaxDynamicSharedMemorySize, (int)shmem);

  ssim3d_tile<<<dim3(TX, TY, NB * TZ), 256, shmem, stream>>>(img1, img2, part);
  ssim3d_reduce<<<1, 256, 0, stream>>>(part, out);
}